// AtomTypeGNN_6863357739280
// MI455X (gfx1250) — compile-verified
//
#include <hip/hip_runtime.h>

// Problem constants (match reference)
#define NN 2048   // atoms
#define FF 64     // dist_exp_size
#define HH 128    // atom_emb_size
#define KK 256    // hidden_size
#define FH (FF * HH)  // 8192 flattened inner GEMM dim

typedef __attribute__((ext_vector_type(2))) float v2f;
typedef __attribute__((ext_vector_type(4))) float v4f;
typedef __attribute__((ext_vector_type(8))) float v8f;

// ---------------------------------------------------------------------------
// CDNA5 async global->LDS copy (ASYNCcnt-tracked, bypasses VGPRs).
//   lds_addr : LDS byte address (low 32 bits of generic pointer)
//   goff     : 32-bit byte offset added to the SGPR base (GVS mode)
//   base     : uniform 64-bit global base (SGPR pair)
// ---------------------------------------------------------------------------
__device__ __forceinline__ void async_copy_b128(uint32_t lds_addr, uint32_t goff,
                                                const float* base) {
  asm volatile("global_load_async_to_lds_b128 %0, %1, %2"
               :
               : "v"(lds_addr), "v"(goff), "s"((unsigned long long)(uintptr_t)base)
               : "memory");
}
__device__ __forceinline__ void wait_async_le2() {
  asm volatile("s_wait_asynccnt 2" ::: "memory");
}
__device__ __forceinline__ void wait_async_0() {
  asm volatile("s_wait_asynccnt 0" ::: "memory");
}

// ---------------------------------------------------------------------------
// Kernel 1: adj_exp[m,f] = sum_n dist_adj[m,n] * dist_exp[m,n,f]
// Streams 1.07 GB of dist_exp -> HBM-bound (~46us floor at 23.3 TB/s).
// One block per row m; each wave covers TWO 256B f-rows per iteration
// (lanes 0-15 -> row n, lanes 16-31 -> row n+1, float4 of f per lane) so
// every lane issues a 16B non-temporal b128 load.
// ---------------------------------------------------------------------------
__global__ __launch_bounds__(256) void adjexp_kernel(
    const float* __restrict__ dist_adj,
    const float* __restrict__ dist_exp,
    float* __restrict__ adj_exp) {
  const int m = blockIdx.x;
  const int tid = threadIdx.x;
  const int wave = tid >> 5;
  const int lane = tid & 31;
  const int half = lane >> 4;      // row parity within the wave's row pair
  const int fq = lane & 15;        // f-quad id (4 floats each)

  const float* __restrict__ arow = dist_adj + (size_t)m * NN;
  const float* __restrict__ ebase = dist_exp + (size_t)m * NN * FF + fq * 4;

  v4f acc = {0.f, 0.f, 0.f, 0.f};
#pragma unroll 2
  for (int n = wave * 2 + half; n < NN; n += 16) {
    const float w = __builtin_nontemporal_load(arow + n);
    const v4f e =
        __builtin_nontemporal_load((const v4f*)(ebase + (size_t)n * FF));
    acc.x = __builtin_fmaf(w, e.x, acc.x);
    acc.y = __builtin_fmaf(w, e.y, acc.y);
    acc.z = __builtin_fmaf(w, e.z, acc.z);
    acc.w = __builtin_fmaf(w, e.w, acc.w);
  }

  __shared__ v4f red[16][16];
  red[wave * 2 + half][fq] = acc;
  __syncthreads();
  if (tid < 16) {
    v4f s = red[0][tid];
#pragma unroll
    for (int j = 1; j < 16; ++j) {
      const v4f r = red[j][tid];
      s.x += r.x; s.y += r.y; s.z += r.z; s.w += r.w;
    }
    *(v4f*)(adj_exp + (size_t)m * FF + tid * 4) = s;
  }
}

// ---------------------------------------------------------------------------
// Kernel 2: feat[n,k] = sum_{f,h} adj_exp[n,f] * W[f,h,k] * emb[n,h]
//   = GEMM X (2048 x 8192) @ W_flat (8192 x 256) with
//     X[n, f*H+h] = adj_exp[n,f]*emb[n,h] formed on the fly in registers.
// fp32 WMMA (V_WMMA_F32_16X16X4_F32): exact fp32 math.
// Block tile 32n x 64k; 8 waves = 2 n-subtiles x 4 k-subtiles.
// W chunks (32 x 64) double-buffered in LDS via async global->LDS b128
// copies overlapped with the WMMA stream (s_wait_asynccnt pipeline).
// Epilogue: softplus then +bias (bias after activation, per reference).
// ---------------------------------------------------------------------------
#define TN 32
#define TK 64
#define TKK 32
#define BPAD 8                       // row stride 72 floats = 288B: 16B-aligned,
                                     // conflict-free half-wave reads
#define ROWB ((TK + BPAD) * 4)       // 288 bytes per Bs row
#define BUFB (TKK * ROWB)            // 9216 bytes per Bs buffer
#define CHB (TKK * KK * 4)           // 32768-byte global stride per chunk
#define NCH (FH / TKK)               // 256 chunks

__global__ __launch_bounds__(256) void feat_kernel(
    const float* __restrict__ adj_exp,
    const float* __restrict__ atom_emb,
    const float* __restrict__ Wfl,   // (F*H, K) row-major == (F,H,K)
    const float* __restrict__ bias,
    float* __restrict__ out) {
  __shared__ float adjS[TN][FF + 1];
  __shared__ float embS[TN][HH + 2];
  __shared__ __attribute__((aligned(16))) float Bs[2][TKK][TK + BPAD];

  const int kbase = blockIdx.x * TK;
  const int nbase = blockIdx.y * TN;
  const int tid = threadIdx.x;

  // --- async B pipeline: per-thread constants (2 x b128 per 32x64 chunk) ---
  const int ra = tid >> 4;           // rows 0..15 (and ra+16 for 2nd copy)
  const int ca = (tid & 15) * 4;     // 16B-aligned column quad
  const uint32_t ldsBs = (uint32_t)(uintptr_t)&Bs[0][0][0];
  const uint32_t lds_a = ldsBs + (uint32_t)(ra * ROWB + ca * 4);
  const uint32_t lds_b = lds_a + 16u * ROWB;
  uint32_t goff_a = (uint32_t)(((ra)*KK + kbase + ca) * 4);
  uint32_t goff_b = goff_a + 16u * KK * 4u;

  // prologue: start chunk 0 into buffer 0
  async_copy_b128(lds_a, goff_a, Wfl);
  async_copy_b128(lds_b, goff_b, Wfl);
  goff_a += CHB;
  goff_b += CHB;

  // --- stage the block's adj_exp (32x64) and atom_emb (32x128) rows once ---
  for (int i4 = tid; i4 < TN * FF / 4; i4 += 256) {
    const int r = i4 / (FF / 4);
    const int c = (i4 % (FF / 4)) * 4;
    const v4f v = *(const v4f*)(adj_exp + (size_t)(nbase + r) * FF + c);
    adjS[r][c + 0] = v.x; adjS[r][c + 1] = v.y;
    adjS[r][c + 2] = v.z; adjS[r][c + 3] = v.w;
  }
  for (int i4 = tid; i4 < TN * HH / 4; i4 += 256) {
    const int r = i4 / (HH / 4);
    const int c = (i4 % (HH / 4)) * 4;
    const v4f v = *(const v4f*)(atom_emb + (size_t)(nbase + r) * HH + c);
    embS[r][c + 0] = v.x; embS[r][c + 1] = v.y;
    embS[r][c + 2] = v.z; embS[r][c + 3] = v.w;
  }

  const int wave = tid >> 5;
  const int lane = tid & 31;
  const int kt = wave & 3;                  // k-subtile 0..3
  const int nt = wave >> 2;                 // n-subtile 0..1
  const int half = lane >> 4;               // ISA half-wave split
  const int arow = (lane & 15) + nt * 16;   // A-matrix row M for this lane
  const int bcol = kt * 16 + (lane & 15);   // B-matrix column N for this lane

  v8f acc = {};

  for (int i = 0; i < NCH; ++i) {
    if (i + 1 < NCH) {
      // prefetch chunk i+1 into the other buffer, then wait for chunk i
      const uint32_t boff = (uint32_t)(((i + 1) & 1) * BUFB);
      async_copy_b128(lds_a + boff, goff_a, Wfl);
      async_copy_b128(lds_b + boff, goff_b, Wfl);
      goff_a += CHB;
      goff_b += CHB;
      wait_async_le2();    // chunk i complete (2 newest ops may remain)
    } else {
      wait_async_0();
    }
    __syncthreads();       // chunk i visible to all waves (also covers adj/emb)

    const int buf = i & 1;
    const int kk0 = i * TKK;
#pragma unroll
    for (int s = 0; s < TKK; s += 4) {
      const int kkb = kk0 + s;
      const int f = kkb >> 7;                 // uniform across the 4 K's
      const int h0 = (kkb & 127) + half * 2;  // lane's two h indices
      // A fragment (16x4 f32): lane (M=arow) holds K = 2*half + {0,1}
      const float a = adjS[arow][f];
      const v2f e = *(const v2f*)&embS[arow][h0];
      v2f A;
      A.x = a * e.x;
      A.y = a * e.y;
      // B fragment (4x16 f32): lane holds rows 2*half + {0,1}, column bcol
      const int r0 = s + half * 2;
      v2f B;
      B.x = Bs[buf][r0 + 0][bcol];
      B.y = Bs[buf][r0 + 1][bcol];
      acc = __builtin_amdgcn_wmma_f32_16x16x4_f32(
          false, A, false, B, (short)0, acc, false, false);
    }
    __syncthreads();       // all reads of buf done before it is re-written
  }

  // Epilogue: softplus(x) + bias  (bias added AFTER activation, per reference)
  const int ocol = kbase + kt * 16 + (lane & 15);
  const float b = bias[ocol];
#pragma unroll
  for (int j = 0; j < 8; ++j) {
    const int orow = nbase + nt * 16 + half * 8 + j;  // C layout: M = 8*half + j
    const float x = acc[j];
    const float sp = fmaxf(x, 0.0f) + log1pf(__expf(-fabsf(x)));
    out[(size_t)orow * KK + ocol] = sp + b;
  }
}

// ---------------------------------------------------------------------------
extern "C" void kernel_launch(void* const* d_in, const int* in_sizes, int n_in,
                              void* d_out, int out_size, void* d_ws,
                              size_t ws_size, hipStream_t stream) {
  const float* dist_adj = (const float*)d_in[0];  // (N,N)
  const float* dist_exp = (const float*)d_in[1];  // (N,N,F)
  const float* atom_emb = (const float*)d_in[2];  // (N,H)
  const float* bilin_w  = (const float*)d_in[3];  // (F,H,K)
  const float* bilin_b  = (const float*)d_in[4];  // (K,)
  float* out = (float*)d_out;                     // (N,K) fp32

  float* adj_exp = (float*)d_ws;                  // N*F*4 = 512 KB scratch

  adjexp_kernel<<<dim3(NN), dim3(256), 0, stream>>>(dist_adj, dist_exp,
                                                    adj_exp);
  feat_kernel<<<dim3(KK / TK, NN / TN), dim3(256), 0, stream>>>(
      adj_exp, atom_emb, bilin_w, bilin_b, out);
}